// LSMGCN_46325517254869
// MI455X (gfx1250) — compile-verified
//
#include <hip/hip_runtime.h>
#include <math.h>

#define N_NODES 50000
#define N_EDGES 800000
#define DIM     256
#define HDIM    128
#define NTILES  (N_NODES / 16)   // 3125, exact
#define WPB     8                // waves (row-tiles) per block in GEMM

typedef __attribute__((ext_vector_type(2))) float v2f;
typedef __attribute__((ext_vector_type(8))) float v8f;

// ---------------- SPMM scatter: y[row] += val * x[col], y pre-zeroed ----------------
template<int D>
__global__ void spmm_scatter_kernel(const int* __restrict__ erow,
                                    const int* __restrict__ ecol,
                                    const float* __restrict__ evalv,
                                    const float* __restrict__ x,
                                    float* __restrict__ y)
{
    constexpr int CHUNK    = 16;            // floats handled per thread
    constexpr int PER_EDGE = D / CHUNK;
    long long tid = (long long)blockIdx.x * blockDim.x + threadIdx.x;
    int e = (int)(tid / PER_EDGE);
    if (e >= N_EDGES) return;
    int c  = (int)(tid % PER_EDGE) * CHUNK;
    int r  = erow[e];
    int cl = ecol[e];
    float v = evalv[e];
    const float4* xs = (const float4*)(x + (size_t)cl * D + c);
    float* yd = y + (size_t)r * D + c;
#pragma unroll
    for (int i = 0; i < CHUNK / 4; ++i) {
        float4 xv = xs[i];
        atomicAdd(yd + i * 4 + 0, v * xv.x);
        atomicAdd(yd + i * 4 + 1, v * xv.y);
        atomicAdd(yd + i * 4 + 2, v * xv.z);
        atomicAdd(yd + i * 4 + 3, v * xv.w);
    }
}

// ---------- P = softmax(Z @ W^T + b); one wave owns a 16-row x 128-col block ----------
// f32 WMMA 16x16x4, wave32 layouts (ISA §7.12.2):
//   A (16x4):  lane = h*16+m; VGPR v holds A[m][kb + 2h + v]
//   B (4x16):  lane = h*16+n; VGPR v holds B[kb + 2h + v][n] = W[col][kb+2h+v]
//   C/D 16x16: lane = h*16+n; VGPR r holds D[r + 8h][n]
// W is staged per 128-k chunk in LDS as wl[k2][col] (k2 = k/2 pairs, 8B each):
// frag read addr = ((kb/2 + h)*128 + col)*8 -> each half reads 128 contiguous B; halves sit 1024 B apart (same banks).
template<int K, bool FUSE_TANH>
__global__ void gemm_softmax_kernel(const float* __restrict__ Z,     // [N, K] (+ slack rows in ws)
                                    const float* __restrict__ W,     // [128, K]
                                    const float* __restrict__ bias,  // [128]
                                    const float* __restrict__ hres,  // [N, 256] (FUSE only)
                                    float* __restrict__ out)         // FUSE: [N,256]; else [N,128]
{
    __shared__ v2f wl[64 * 128];                // 64 KB

    int wave = threadIdx.x >> 5;
    int lane = threadIdx.x & 31;
    int tile = blockIdx.x * WPB + wave;
    int rowBase = tile * 16;
    int half = lane >> 4;                       // 0/1
    int ln   = lane & 15;                       // 0..15

    v8f acc[8];
#pragma unroll
    for (int t = 0; t < 8; ++t)
#pragma unroll
        for (int r = 0; r < 8; ++r) acc[t][r] = 0.0f;

    const float* zrow = Z + (size_t)(rowBase + ln) * K + 2 * half;

    constexpr int NCHUNK = K / 128;
    for (int ch = 0; ch < NCHUNK; ++ch) {
        const int k0 = ch * 128;
        __syncthreads();                        // protect wl readers of previous chunk
        // stage W[:, k0:k0+128] -> wl[k2][col]; col fastest keeps ds_store conflict-free
        for (int i = threadIdx.x; i < 64 * 128; i += WPB * 32) {
            int k2  = i >> 7;
            int col = i & 127;
            wl[i] = *(const v2f*)(W + (size_t)col * K + k0 + 2 * k2);
        }
        __syncthreads();

        const float* zp = zrow + k0;
        v2f a_cur = *(const v2f*)(zp);
        for (int kb = 0; kb < 128; kb += 4) {
            const v2f* wrow = &wl[((kb >> 1) + half) * 128 + ln];
            v2f bf[8];
#pragma unroll
            for (int t = 0; t < 8; ++t) bf[t] = wrow[t * 16];
            v2f a_next = a_cur;
            if (kb + 4 < 128) a_next = *(const v2f*)(zp + kb + 4);
#pragma unroll
            for (int t = 0; t < 8; ++t)
                acc[t] = __builtin_amdgcn_wmma_f32_16x16x4_f32(
                    false, a_cur, false, bf[t], (short)0, acc[t], false, false);
            a_cur = a_next;
        }
    }

    // bias (column t*16+ln, identical across the 8 rows this lane holds)
#pragma unroll
    for (int t = 0; t < 8; ++t) {
        float bv = bias[t * 16 + ln];
#pragma unroll
        for (int r = 0; r < 8; ++r) acc[t][r] += bv;
    }

    // per-row softmax across 128 cols: row m = r + 8*half lives in the 16 lanes
    // of this half across acc[0..7][r] -> reduce locally, then shfl_xor 1,2,4,8.
#pragma unroll
    for (int r = 0; r < 8; ++r) {
        float mx = acc[0][r];
#pragma unroll
        for (int t = 1; t < 8; ++t) mx = fmaxf(mx, acc[t][r]);
        for (int s = 1; s < 16; s <<= 1) mx = fmaxf(mx, __shfl_xor(mx, s, 32));
        float sum = 0.0f;
#pragma unroll
        for (int t = 0; t < 8; ++t) {
            float ev = __expf(acc[t][r] - mx);
            acc[t][r] = ev;
            sum += ev;
        }
        for (int s = 1; s < 16; s <<= 1) sum += __shfl_xor(sum, s, 32);
        float inv = 1.0f / sum;
#pragma unroll
        for (int t = 0; t < 8; ++t) acc[t][r] *= inv;
    }

    if (tile < NTILES) {
        // 16 consecutive lanes of a half write 16 consecutive floats of one row
#pragma unroll
        for (int r = 0; r < 8; ++r) {
            int m = rowBase + r + 8 * half;
#pragma unroll
            for (int t = 0; t < 8; ++t) {
                int c = t * 16 + ln;
                if (FUSE_TANH) {
                    size_t o = (size_t)m * 256 + 128 + c;
                    out[o] = tanhf(acc[t][r] + hres[o]);
                } else {
                    out[(size_t)m * 128 + c] = acc[t][r];
                }
            }
        }
    }
}

// ---------------- out[:, :128] = tanh(AxW0 + h[:, :128]) ----------------
__global__ void final_lo_kernel(const float* __restrict__ axw0,
                                const float* __restrict__ hres,
                                float* __restrict__ out)
{
    int i = blockIdx.x * blockDim.x + threadIdx.x;
    if (i >= N_NODES * HDIM) return;
    int m = i >> 7;
    int c = i & 127;
    out[(size_t)m * 256 + c] = tanhf(axw0[i] + hres[(size_t)m * 256 + c]);
}

extern "C" void kernel_launch(void* const* d_in, const int* in_sizes, int n_in,
                              void* d_out, int out_size, void* d_ws, size_t ws_size,
                              hipStream_t stream) {
    (void)in_sizes; (void)n_in; (void)out_size; (void)ws_size;
    const float* X   = (const float*)d_in[0];   // inputs_emb [N,256]
    const float* W0  = (const float*)d_in[1];   // [128,256]
    const float* b0  = (const float*)d_in[2];   // [128]
    const float* W1  = (const float*)d_in[3];   // [128,128]
    const float* b1  = (const float*)d_in[4];   // [128]
    const float* ev  = (const float*)d_in[5];   // [E]
    const int*   er  = (const int*)d_in[6];     // [E]
    const int*   ec  = (const int*)d_in[7];     // [E]
    float* out = (float*)d_out;                 // [N,256]

    float* h    = (float*)d_ws;                 // [N,256]
    float* z    = h + (size_t)N_NODES * DIM;    // [N,256] (reused as [N,128])
    float* axw0 = z + (size_t)N_NODES * DIM;    // [N,128]

    const int TPB = 256;
    long long t256 = (long long)N_EDGES * (DIM / 16);
    long long t128 = (long long)N_EDGES * (HDIM / 16);
    int g256 = (int)((t256 + TPB - 1) / TPB);
    int g128 = (int)((t128 + TPB - 1) / TPB);
    int ggemm = (NTILES + WPB - 1) / WPB;       // 8 row-tiles (waves) per block

    // h = A @ X
    hipMemsetAsync(h, 0, (size_t)N_NODES * DIM * sizeof(float), stream);
    spmm_scatter_kernel<DIM><<<g256, TPB, 0, stream>>>(er, ec, ev, X, h);

    // z0 = A @ h ; AxW0 = softmax(z0 @ W0^T + b0)
    hipMemsetAsync(z, 0, (size_t)N_NODES * DIM * sizeof(float), stream);
    spmm_scatter_kernel<DIM><<<g256, TPB, 0, stream>>>(er, ec, ev, h, z);
    gemm_softmax_kernel<DIM, false><<<ggemm, WPB * 32, 0, stream>>>(z, W0, b0, nullptr, axw0);

    // z1 = A @ AxW0 ; out[:,128:] = tanh(softmax(z1 @ W1^T + b1) + h[:,128:])
    hipMemsetAsync(z, 0, (size_t)N_NODES * HDIM * sizeof(float), stream);
    spmm_scatter_kernel<HDIM><<<g128, TPB, 0, stream>>>(er, ec, ev, axw0, z);
    gemm_softmax_kernel<HDIM, true><<<ggemm, WPB * 32, 0, stream>>>(z, W1, b1, h, out);

    // out[:, :128] = tanh(AxW0 + h[:, :128])
    final_lo_kernel<<<(N_NODES * HDIM + TPB - 1) / TPB, TPB, 0, stream>>>(axw0, h, out);
}